// LIF_69707319214329
// MI455X (gfx1250) — compile-verified
//
#include <hip/hip_runtime.h>

typedef float v2f __attribute__((ext_vector_type(2)));
typedef float v4f __attribute__((ext_vector_type(4)));
typedef float v8f __attribute__((ext_vector_type(8)));

constexpr int   IN_F   = 8192;
constexpr int   OUT_F  = 8192;
constexpr int   KDIM   = 1024;
constexpr float DECAYF = 0.99004983374916811f;  // exp(-0.01/1.0)
constexpr float THRESH = 20.0f;

// ---------------------------------------------------------------------------
// Phase 1: rowsum(x) via V_WMMA_F32_16X16X4_F32 with B = ones, then LIF
// threshold -> spikes vector in d_ws.
//
// Block = 128 threads = 4 waves. Each block owns 16 rows of x; each wave
// accumulates a 256-element K-chunk of those rows with WMMA (A = 16x4 slice
// of x, B = ones => D[m,n] = partial rowsum(m)). Partials combined in LDS.
//
// A-matrix f32 16x4 VGPR layout (ISA 7.12.2): lanes 0-15 hold K=0,1 in
// VGPR0,1; lanes 16-31 hold K=2,3. So lane (m = lane&15) loads the float2 at
// x[rowBase+m][k + 2*(lane>>4)].
// C/D f32 16x16 layout: VGPR g, lane L -> M = g + (L>=16 ? 8 : 0), N = L%16.
// => lane 0 holds rowsums M=0..7 in acc[0..7]; lane 16 holds M=8..15.
// ---------------------------------------------------------------------------
__global__ __launch_bounds__(128) void lif_spikes_kernel(
    const float* __restrict__ x, const float* __restrict__ v,
    const float* __restrict__ spikes_in, float* __restrict__ spikes_out)
{
    __shared__ float partial[4][16];

    const int wave    = threadIdx.x >> 5;         // 0..3
    const int lane    = threadIdx.x & 31;
    const int m       = lane & 15;                // row within tile
    const int kOff    = (lane >> 4) << 1;         // 0 or 2 (A layout)
    const int rowBase = blockIdx.x * 16;
    const int kBase   = wave * (KDIM / 4);        // 256-wide K chunk per wave

    const float* rowPtr = x + (size_t)(rowBase + m) * KDIM + kBase + kOff;

    v8f acc0 = {};
    v8f acc1 = {};
    const v2f bOnes = {1.0f, 1.0f};

    // 256 K-values per wave, 4 per WMMA, 2 independent accumulator chains.
    for (int k = 0; k < KDIM / 4; k += 8) {
        v2f a0 = *(const v2f*)(rowPtr + k);
        v2f a1 = *(const v2f*)(rowPtr + k + 4);
        acc0 = __builtin_amdgcn_wmma_f32_16x16x4_f32(
            false, a0, false, bOnes, (short)0, acc0, false, false);
        acc1 = __builtin_amdgcn_wmma_f32_16x16x4_f32(
            false, a1, false, bOnes, (short)0, acc1, false, false);
    }
    v8f acc = acc0 + acc1;

    // Lanes with N==0 (lane 0 and lane 16) each hold 8 rowsums.
    if (m == 0) {
        const int mBase = (lane >> 4) * 8;
        #pragma unroll
        for (int g = 0; g < 8; ++g)
            partial[wave][mBase + g] = acc[g];
    }
    __syncthreads();

    if (threadIdx.x < 16) {
        const int   row    = rowBase + threadIdx.x;
        const float rowsum = partial[0][threadIdx.x] + partial[1][threadIdx.x]
                           + partial[2][threadIdx.x] + partial[3][threadIdx.x];
        const float vn = v[row] * DECAYF + rowsum * 10.0f;
        spikes_out[row] = (vn >= THRESH) ? 1.0f : spikes_in[row];
    }
}

// ---------------------------------------------------------------------------
// Phase 2: out[o,i] = spikes[i] * weight[o,i]  (520 MB HBM stream).
// float4 coalesced, non-temporal on the 512 MB weight/out stream so the
// 32 KB spikes vector stays cache-resident.
// ---------------------------------------------------------------------------
__global__ __launch_bounds__(256) void spike_scale_kernel(
    const float* __restrict__ weight, const float* __restrict__ spikes,
    float* __restrict__ out)
{
    const size_t i = (size_t)blockIdx.x * blockDim.x + threadIdx.x;

    const v4f* __restrict__ w4 = (const v4f*)weight;
    const v4f* __restrict__ s4 = (const v4f*)spikes;
    v4f*       __restrict__ o4 = (v4f*)out;

    const int col4 = (int)(i & (size_t)(IN_F / 4 - 1));

    v4f w = __builtin_nontemporal_load(w4 + i);   // streamed once
    v4f s = s4[col4];                             // 32 KB, cache-resident
    __builtin_nontemporal_store(w * s, o4 + i);   // streamed once
}

extern "C" void kernel_launch(void* const* d_in, const int* in_sizes, int n_in,
                              void* d_out, int out_size, void* d_ws, size_t ws_size,
                              hipStream_t stream) {
    (void)in_sizes; (void)n_in; (void)out_size; (void)ws_size;

    const float* x       = (const float*)d_in[0];  // [8192, 1024]
    const float* weight  = (const float*)d_in[1];  // [8192, 8192]
    const float* v       = (const float*)d_in[2];  // [8192]
    const float* spikes  = (const float*)d_in[3];  // [8192]
    // d_in[4] = t (unused)

    float* out       = (float*)d_out;              // [8192, 8192]
    float* spikes_ws = (float*)d_ws;               // [8192] scratch

    // Phase 1: 8192 rows / 16 rows-per-block = 512 blocks of 128 threads.
    lif_spikes_kernel<<<IN_F / 16, 128, 0, stream>>>(x, v, spikes, spikes_ws);

    // Phase 2: 8192*8192/4 float4 elements, 256 threads/block.
    const size_t n4 = (size_t)OUT_F * IN_F / 4;    // 16,777,216
    spike_scale_kernel<<<(unsigned)(n4 / 256), 256, 0, stream>>>(
        weight, spikes_ws, out);
}